// RoIAlign_73117523247233
// MI455X (gfx1250) — compile-verified
//
#include <hip/hip_runtime.h>

// 3D RoI Align, forward only.
// features: (B=2, C=128, H=64, W=64, L=32) f32, rois: (N,7) f32, scale: f32[1]
// out: (N, C, 7, 7, 7) f32
//
// Memory-bound: ~173 MB of traffic vs ~0.16 GFLOP -> L2/HBM gather kernel.
// WMMA densification would inflate FLOPs ~70x (f32 16x16x4 is the slowest
// WMMA dtype) and save zero bytes, so the vector gather path is optimal.
//
// v2: point decode hoisted out of the hot loop (one decode per thread, not
// per output), inner loop = channel walk with immediate-offset b64 gathers,
// non-temporal stores keep the write-once output from evicting the
// L2-resident feature tensor (128 MB of 192 MB L2).

#define AH 7
#define AW 7
#define AL 7
#define NPTS (AH * AW * AL)   // 343
#define C_   128
#define H_   64
#define W_   64
#define L_   32
#define CSTRIDE (H_ * W_ * L_)   // elements between channels (131072)
#define CTILE   16
#define CGROUPS (C_ / CTILE)     // 8 channel-groups per RoI

// 4-byte-aligned pair load: l0 can be odd so the address is only dword
// aligned; gfx1250 DWORD alignment mode allows b64 loads at 4B alignment and
// the backend emits a single global_load_b64.
struct F2 { float lo, hi; };

__global__ __launch_bounds__(352)
void roi_align3d_fwd(const float* __restrict__ feat,
                     const float* __restrict__ rois,
                     const float* __restrict__ scale_ptr,
                     float* __restrict__ out,
                     int N)
{
    __shared__ int   s_base[3][8];   // [axis h/w/l][sample 0..6]
    __shared__ float s_frac[3][8];
    __shared__ int   s_b;

    const int blk = blockIdx.x;
    const int n   = blk / CGROUPS;     // RoI index
    const int cg  = blk % CGROUPS;     // channel group
    const int tid = threadIdx.x;
    if (n >= N) return;

    // --- per-RoI sample metadata, computed once per block into LDS ---------
    if (tid < 3 * AH + 1) {            // 22 threads
        const float  scale = *scale_ptr;
        const float* r     = rois + (size_t)n * 7;
        if (tid == 3 * AH) {
            s_b = (int)r[0];
        } else {
            const int axis = tid / AH;          // 0=h, 1=w, 2=l
            const int i    = tid % AH;
            const float lo   = r[1 + axis] * scale;
            const float hi   = r[4 + axis] * scale;
            const float len  = fmaxf(hi - lo + 1.0f, 0.0f);
            const float step = len / (float)(AH - 1);
            const int dimmax = (axis == 0 ? H_ : (axis == 1 ? W_ : L_)) - 1;
            float g = lo + step * (float)i;
            g = fminf(fmaxf(g, 0.0f), (float)dimmax);
            int b0 = (int)floorf(g);
            // clamped-base lerp: base <= dim-2 so base+1 is always in range;
            // frac = g-base reproduces the reference's clamped corner math
            // exactly (floor(g)==dim-1 => frac==1 => picks f[dim-1]).
            if (b0 > dimmax - 1) b0 = dimmax - 1;
            s_base[axis][i] = b0;
            s_frac[axis][i] = g - (float)b0;
        }
    }
    __syncthreads();   // CDNA5 split barrier

    const int p = tid;                 // one sample point per thread
    if (p >= NPTS) return;

    // --- one-time decode: p -> (ih, iw, il) -> gather offset + weights -----
    const int ih = p / (AW * AL);
    const int r2 = p - ih * (AW * AL);
    const int iw = r2 / AL;
    const int il = r2 - iw * AL;

    const int   h0 = s_base[0][ih]; const float fh = s_frac[0][ih];
    const int   w0 = s_base[1][iw]; const float fw = s_frac[1][iw];
    const int   l0 = s_base[2][il]; const float fl = s_frac[2][il];

    const int b  = s_b;
    const int c0 = cg * CTILE;

    const float* pp = feat + (size_t)(b * C_ + c0) * (size_t)CSTRIDE
                           + (size_t)((h0 * W_ + w0) * L_ + l0);
    float*       ob = out  + ((size_t)n * C_ + c0) * (size_t)NPTS + p;

    // --- hot loop: walk 16 channels; body = 4 b64 gathers + 7 FMA + 1 store -
#pragma unroll 4
    for (int cl = 0; cl < CTILE; ++cl) {
        const F2 v00 = *(const F2*)(pp);                  // (h0,  w0,  l0..+1)
        const F2 v01 = *(const F2*)(pp + L_);             // (h0,  w0+1)
        const F2 v10 = *(const F2*)(pp + W_ * L_);        // (h0+1, w0 )
        const F2 v11 = *(const F2*)(pp + W_ * L_ + L_);   // (h0+1, w0+1)

        const float a00 = fmaf(fl, v00.hi - v00.lo, v00.lo);
        const float a01 = fmaf(fl, v01.hi - v01.lo, v01.lo);
        const float a10 = fmaf(fl, v10.hi - v10.lo, v10.lo);
        const float a11 = fmaf(fl, v11.hi - v11.lo, v11.lo);
        const float a0  = fmaf(fw, a01 - a00, a00);
        const float a1  = fmaf(fw, a11 - a10, a10);
        const float res = fmaf(fh, a1 - a0, a0);

        // write-once output: stream past L2 (TH=NT) so the 128 MB feature
        // tensor stays L2-resident
        __builtin_nontemporal_store(res, ob);

        pp += (size_t)CSTRIDE;
        ob += NPTS;
    }
}

extern "C" void kernel_launch(void* const* d_in, const int* in_sizes, int n_in,
                              void* d_out, int out_size, void* d_ws, size_t ws_size,
                              hipStream_t stream) {
    (void)n_in; (void)out_size; (void)d_ws; (void)ws_size;
    const float* feat  = (const float*)d_in[0];
    const float* rois  = (const float*)d_in[1];
    const float* scale = (const float*)d_in[2];
    float*       out   = (float*)d_out;

    const int N = in_sizes[1] / 7;                 // 256 RoIs
    dim3 grid((unsigned)(N * CGROUPS));            // 2048 blocks
    dim3 block(352);                               // 11 wave32s, 343 active
    roi_align3d_fwd<<<grid, block, 0, stream>>>(feat, rois, scale, out, N);
}